// Hetero6Layer_23124103921911
// MI455X (gfx1250) — compile-verified
//
#include <hip/hip_runtime.h>

#define HDIM 128

typedef __attribute__((ext_vector_type(2))) float v2f;
typedef __attribute__((ext_vector_type(8))) float v8f;

// ---------------------------------------------------------------------------
// Sum up to 4 weight matrices [H,H] and 4 biases [H] into combined Wc, bc.
// ---------------------------------------------------------------------------
__global__ void combine_wb(const float* __restrict__ W0, const float* __restrict__ W1,
                           const float* __restrict__ W2, const float* __restrict__ W3,
                           const float* __restrict__ b0, const float* __restrict__ b1,
                           const float* __restrict__ b2, const float* __restrict__ b3,
                           float* __restrict__ Wc, float* __restrict__ bc) {
    int i = blockIdx.x * blockDim.x + threadIdx.x;
    if (i < HDIM * HDIM) {
        float v = W0[i] + W1[i];
        if (W2) v += W2[i];
        if (W3) v += W3[i];
        Wc[i] = v;
    }
    if (i < HDIM) {
        float v = b0[i] + b1[i];
        if (b2) v += b2[i];
        if (b3) v += b3[i];
        bc[i] = v;
    }
}

// ---------------------------------------------------------------------------
// Y[M,128] = X[M,128] @ W[128,128] (+ bias), fp32 WMMA 16x16x4.
// 256 threads = 8 waves; each wave produces a 16-row x 128-col strip.
// W is staged in LDS K-pair interleaved: sW2[p*128+col] = {W[2p][col],
// W[2p+1][col]} so each B fragment is ONE aligned ds_load_b64 straight into
// an even VGPR pair (no v_mov shuffling before the WMMAs). The X tile is
// staged row-major; each A fragment is one aligned b64 LDS load.
// All waves keep EXEC all-ones through the WMMAs (ISA requirement);
// out-of-range rows are zero-filled on load and guarded on store.
// ---------------------------------------------------------------------------
__launch_bounds__(256, 2)
__global__ void gemm128_wmma(const float* __restrict__ X,
                             const float* __restrict__ W,
                             const float* __restrict__ bias,   // may be null
                             float* __restrict__ Y,
                             int M) {
    __shared__ v2f   sW2[64 * HDIM];    // 64 KB, K-pair interleaved
    __shared__ float sX[128 * HDIM];    // 64 KB
    __shared__ float sB[HDIM];

    const int  tid       = threadIdx.x;
    const long blockRow0 = (long)blockIdx.x * 128;

    // stage weights K-pair interleaved: two coalesced float2 row reads per step
    for (int i = tid * 2; i < 64 * HDIM; i += 256 * 2) {
        int p   = i >> 7;        // k-pair index
        int col = i & 127;
        float2 r0 = *(const float2*)(W + (2 * p) * HDIM + col);
        float2 r1 = *(const float2*)(W + (2 * p + 1) * HDIM + col);
        v2f w0; w0.x = r0.x; w0.y = r1.x;
        v2f w1; w1.x = r0.y; w1.y = r1.y;
        sW2[i]     = w0;
        sW2[i + 1] = w1;
    }

    // stage 128 rows of X; rows >= M become zero
    for (int i = tid * 4; i < 128 * HDIM; i += 256 * 4) {
        long r = blockRow0 + (i >> 7);
        float4 v;
        if (r < M) v = *(const float4*)(X + blockRow0 * HDIM + i);
        else       v = make_float4(0.f, 0.f, 0.f, 0.f);
        *(float4*)(sX + i) = v;
    }
    if (tid < HDIM) sB[tid] = bias ? bias[tid] : 0.0f;
    __syncthreads();

    const int    wave = tid >> 5;        // 0..7 -> 16 rows each
    const int    lane = tid & 31;
    const int    l    = lane & 15;       // row-in-tile / col-in-tile
    const int    hi   = lane >> 4;       // selects K pair {0,1} vs {2,3}
    const float* xw   = sX + wave * 16 * HDIM;

    // Accumulators: 8 N-tiles of 16x16. C/D layout: VGPR v -> row (v + 8*hi),
    // lane%16 -> column. Initialize with bias broadcast down rows.
    v8f acc[8];
#pragma unroll
    for (int n = 0; n < 8; ++n) {
        float b = sB[n * 16 + l];
#pragma unroll
        for (int v = 0; v < 8; ++v) acc[n][v] = b;
    }

#pragma unroll 4
    for (int k0 = 0; k0 < HDIM; k0 += 4) {
        const int kp = (k0 >> 1) + hi;   // this half-wave's k-pair
        // A frag (16x4 fp32): lane(l,hi) holds row l, K = {2kp, 2kp+1}
        v2f a = *(const v2f*)(xw + l * HDIM + (kp << 1));
#pragma unroll
        for (int n = 0; n < 8; ++n) {
            // B frag (4x16 fp32): {W[2kp][col], W[2kp+1][col]} in one b64
            v2f b = sW2[kp * HDIM + n * 16 + l];
            acc[n] = __builtin_amdgcn_wmma_f32_16x16x4_f32(
                false, a, false, b, (short)0, acc[n], false, false);
        }
    }

    // store: row = blockRow0 + wave*16 + hi*8 + v, col = n*16 + l
    const long row0 = blockRow0 + wave * 16 + hi * 8;
#pragma unroll
    for (int v = 0; v < 8; ++v) {
        long r = row0 + v;
        if (r < M) {
            float* yp = Y + r * HDIM + l;
#pragma unroll
            for (int n = 0; n < 8; ++n)
                yp[n * 16] = acc[n][v];
        }
    }
}

// ---------------------------------------------------------------------------
// One wave per edge: out[dst] += ew * msg[src]. 32 lanes x float4 = 128 cols.
// ---------------------------------------------------------------------------
__launch_bounds__(256)
__global__ void edge_scatter(const int* __restrict__ ei,    // [2][E]: src, dst
                             const float* __restrict__ ew,  // [E]
                             const float* __restrict__ msg, // [n_src][128]
                             float* __restrict__ out,       // dst-type base
                             int nE) {
    int e = blockIdx.x * 8 + (threadIdx.x >> 5);
    if (e >= nE) return;
    const int    lane = threadIdx.x & 31;
    const int    s    = ei[e];
    const int    d    = ei[nE + e];
    const float  w    = ew[e];
    const float4 m    = *(const float4*)(msg + (long)s * HDIM + lane * 4);
    float*       o    = out + (long)d * HDIM + lane * 4;
    atomicAdd(o + 0, m.x * w);
    atomicAdd(o + 1, m.y * w);
    atomicAdd(o + 2, m.z * w);
    atomicAdd(o + 3, m.w * w);
}

// ---------------------------------------------------------------------------
extern "C" void kernel_launch(void* const* d_in, const int* in_sizes, int n_in,
                              void* d_out, int out_size, void* d_ws, size_t ws_size,
                              hipStream_t stream) {
    const float* x_app = (const float*)d_in[0];
    const float* x_sys = (const float*)d_in[1];
    const float* x_bnd = (const float*)d_in[2];
    const float* x_cmp = (const float*)d_in[3];

    const int*   ei_app_sys = (const int*)d_in[4];   const float* ew_app_sys = (const float*)d_in[5];
    const int*   ei_sys_app = (const int*)d_in[6];   const float* ew_sys_app = (const float*)d_in[7];
    const int*   ei_app_bnd = (const int*)d_in[8];   const float* ew_app_bnd = (const float*)d_in[9];
    const int*   ei_bnd_app = (const int*)d_in[10];  const float* ew_bnd_app = (const float*)d_in[11];
    const int*   ei_app_cmp = (const int*)d_in[12];  const float* ew_app_cmp = (const float*)d_in[13];
    const int*   ei_cmp_app = (const int*)d_in[14];  const float* ew_cmp_app = (const float*)d_in[15];

    const float* W_app = (const float*)d_in[16];  const float* b_app = (const float*)d_in[17];
    const float* W_sys = (const float*)d_in[18];  const float* b_sys = (const float*)d_in[19];
    const float* W_bnd = (const float*)d_in[20];  const float* b_bnd = (const float*)d_in[21];
    const float* W_cmp = (const float*)d_in[22];  const float* b_cmp = (const float*)d_in[23];

    const float* Wn_app_sys = (const float*)d_in[24]; const float* Wd_app_sys = (const float*)d_in[25]; const float* bd_app_sys = (const float*)d_in[26];
    const float* Wn_sys_app = (const float*)d_in[27]; const float* Wd_sys_app = (const float*)d_in[28]; const float* bd_sys_app = (const float*)d_in[29];
    const float* Wn_app_bnd = (const float*)d_in[30]; const float* Wd_app_bnd = (const float*)d_in[31]; const float* bd_app_bnd = (const float*)d_in[32];
    const float* Wn_bnd_app = (const float*)d_in[33]; const float* Wd_bnd_app = (const float*)d_in[34]; const float* bd_bnd_app = (const float*)d_in[35];
    const float* Wn_app_cmp = (const float*)d_in[36]; const float* Wd_app_cmp = (const float*)d_in[37]; const float* bd_app_cmp = (const float*)d_in[38];
    const float* Wn_cmp_app = (const float*)d_in[39]; const float* Wd_cmp_app = (const float*)d_in[40]; const float* bd_cmp_app = (const float*)d_in[41];

    const int NAPP = 100000, NSYS = 50000, NBND = 50000, NCMP = 50000, NE = 400000;

    float* out_app = (float*)d_out;
    float* out_sys = out_app + (long)NAPP * HDIM;
    float* out_bnd = out_sys + (long)NSYS * HDIM;
    float* out_cmp = out_bnd + (long)NBND * HDIM;

    // workspace: msg buffer (100000*128 f32 = 51.2MB) + 4 combined W + 4 biases
    float* ws  = (float*)d_ws;
    float* msg = ws;
    float* Wc  = ws + (long)NAPP * HDIM;
    float* bc  = Wc + 4 * HDIM * HDIM;

    dim3 blk(256);

    // Fold all dense contributions per destination type into one W/b.
    combine_wb<<<64, blk, 0, stream>>>(W_app, Wd_sys_app, Wd_bnd_app, Wd_cmp_app,
                                       b_app, bd_sys_app, bd_bnd_app, bd_cmp_app,
                                       Wc + 0 * HDIM * HDIM, bc + 0 * HDIM);
    combine_wb<<<64, blk, 0, stream>>>(W_sys, Wd_app_sys, nullptr, nullptr,
                                       b_sys, bd_app_sys, nullptr, nullptr,
                                       Wc + 1 * HDIM * HDIM, bc + 1 * HDIM);
    combine_wb<<<64, blk, 0, stream>>>(W_bnd, Wd_app_bnd, nullptr, nullptr,
                                       b_bnd, bd_app_bnd, nullptr, nullptr,
                                       Wc + 2 * HDIM * HDIM, bc + 2 * HDIM);
    combine_wb<<<64, blk, 0, stream>>>(W_cmp, Wd_app_cmp, nullptr, nullptr,
                                       b_cmp, bd_app_cmp, nullptr, nullptr,
                                       Wc + 3 * HDIM * HDIM, bc + 3 * HDIM);

    auto nblocks = [](int m) { return (m + 127) / 128; };

    // Dense part written directly to d_out (fully overwrites every call).
    gemm128_wmma<<<nblocks(NAPP), blk, 0, stream>>>(x_app, Wc + 0 * HDIM * HDIM, bc + 0 * HDIM, out_app, NAPP);
    gemm128_wmma<<<nblocks(NSYS), blk, 0, stream>>>(x_sys, Wc + 1 * HDIM * HDIM, bc + 1 * HDIM, out_sys, NSYS);
    gemm128_wmma<<<nblocks(NBND), blk, 0, stream>>>(x_bnd, Wc + 2 * HDIM * HDIM, bc + 2 * HDIM, out_bnd, NBND);
    gemm128_wmma<<<nblocks(NCMP), blk, 0, stream>>>(x_cmp, Wc + 3 * HDIM * HDIM, bc + 3 * HDIM, out_cmp, NCMP);

    const int sblocks = (NE + 7) / 8;   // one wave per edge, 8 edges / block

    // Per relation: msg = x_src @ Wn (reused buffer; stream order serializes),
    // then scatter-add into destination block of d_out.
    gemm128_wmma<<<nblocks(NAPP), blk, 0, stream>>>(x_app, Wn_app_sys, nullptr, msg, NAPP);
    edge_scatter<<<sblocks, blk, 0, stream>>>(ei_app_sys, ew_app_sys, msg, out_sys, NE);

    gemm128_wmma<<<nblocks(NSYS), blk, 0, stream>>>(x_sys, Wn_sys_app, nullptr, msg, NSYS);
    edge_scatter<<<sblocks, blk, 0, stream>>>(ei_sys_app, ew_sys_app, msg, out_app, NE);

    gemm128_wmma<<<nblocks(NAPP), blk, 0, stream>>>(x_app, Wn_app_bnd, nullptr, msg, NAPP);
    edge_scatter<<<sblocks, blk, 0, stream>>>(ei_app_bnd, ew_app_bnd, msg, out_bnd, NE);

    gemm128_wmma<<<nblocks(NBND), blk, 0, stream>>>(x_bnd, Wn_bnd_app, nullptr, msg, NBND);
    edge_scatter<<<sblocks, blk, 0, stream>>>(ei_bnd_app, ew_bnd_app, msg, out_app, NE);

    gemm128_wmma<<<nblocks(NAPP), blk, 0, stream>>>(x_app, Wn_app_cmp, nullptr, msg, NAPP);
    edge_scatter<<<sblocks, blk, 0, stream>>>(ei_app_cmp, ew_app_cmp, msg, out_cmp, NE);

    gemm128_wmma<<<nblocks(NCMP), blk, 0, stream>>>(x_cmp, Wn_cmp_app, nullptr, msg, NCMP);
    edge_scatter<<<sblocks, blk, 0, stream>>>(ei_cmp_app, ew_cmp_app, msg, out_app, NE);
}